// Spline_63977832841689
// MI455X (gfx1250) — compile-verified
//
#include <hip/hip_runtime.h>
#include <hip/hip_bf16.h>

typedef __attribute__((ext_vector_type(2))) float v2f;
typedef __attribute__((ext_vector_type(8))) float v8f;

#define SPL_W    8192
#define SPL_B    512
#define SPL_R    4
#define TILE     256
#define YS       308   // y halo row stride (need 307)
#define BS       306   // rhs row stride (need 305), even for 8B-aligned v2f loads
#define MS       277   // M row stride (need 272)
#define NINNER   (SPL_W - 4)

// rho = sqrt(3)-2 root of r^2+4r+1=0 ; C = 1/(2*sqrt(3))
#define RHO   (-0.26794919243112270647f)
#define CINV  (0.28867513459481288225f)

#if __has_builtin(__builtin_amdgcn_wmma_f32_16x16x4_f32)
#define HAVE_WMMA_F32X4 1
#else
#define HAVE_WMMA_F32X4 0
#endif

__global__ __launch_bounds__(256) void Spline_63977832841689_kernel(
    const float* __restrict__ x, float* __restrict__ out) {
  const int tid  = threadIdx.x;
  const int c0   = blockIdx.x * TILE;   // first interval column of this block
  const int row0 = blockIdx.y * 16;     // first batch row of this block

  __shared__ float y_s[16 * YS];
  __shared__ float b_s[16 * BS];
  __shared__ float m_s[16 * MS];

  // ---- Phase 1: stage y halo: global cols [c0-17, c0+289] (clamped) ----
  for (int idx = tid; idx < 16 * 307; idx += 256) {
    int rr = idx / 307, cc = idx - rr * 307;
    int gc = c0 - 17 + cc;
    gc = gc < 0 ? 0 : (gc > SPL_W - 1 ? SPL_W - 1 : gc);
    const float* p = &x[(size_t)(row0 + rr) * SPL_W + gc];
    __builtin_prefetch(p + 4096, 0, 0);   // speculative, gfx1250 global_prefetch_b8
    y_s[rr * YS + cc] = *p;
  }
  __syncthreads();

  // ---- Phase 2: RHS b_j = d_{j+2} (+ not-a-knot end adjustments), zero-padded ----
  // b_s[rr][jj] holds b for inner index j = c0 - 18 + jj ; y_{j+1} sits at y_s[rr][jj]
  for (int idx = tid; idx < 16 * 305; idx += 256) {
    int rr = idx / 305, jj = idx - rr * 305;
    int j = c0 - 18 + jj;
    float bv = 0.f;
    if (j >= 0 && j <= NINNER - 1) {
      const float* yr = &y_s[rr * YS];
      bv = 6.f * (yr[jj + 2] - 2.f * yr[jj + 1] + yr[jj]);
      if (j == 0) {                      // b_0 -= M1  (only happens in first block)
        bv -= (yr[19] - 2.f * yr[18] + yr[17]);
      }
      if (j == NINNER - 1) {             // b_{n-1} -= M_{w-2} (only in last block)
        int ie = SPL_W - 1 - c0 + 17;
        bv -= (yr[ie] - 2.f * yr[ie - 1] + yr[ie - 2]);
      }
    }
    b_s[rr * BS + jj] = bv;
  }
  __syncthreads();

  // ---- Phase 3: M tile = Toeplitz-inverse convolution (33 taps of rho^|t|) ----
  const int lane = tid & 31;
  const int wave = tid >> 5;
#if HAVE_WMMA_F32X4
  {
    const int nn  = lane & 15;            // B/D column within subtile
    const int klo = (lane >> 4) << 1;     // K pair: {0,1} or {2,3}
    // Constant banded weight fragments: B_c[k][n] = rho^|4c+k-n-16| (0 outside band)
    v2f Bw[12];
#pragma unroll
    for (int c = 0; c < 12; ++c) {
      float wv0 = 0.f, wv1 = 0.f;
#pragma unroll
      for (int dk = 0; dk < 2; ++dk) {
        int t = 4 * c + klo + dk - nn - 16;
        int m = t < 0 ? -t : t;
        float p = 0.f;
        if (m <= 16) { p = 1.f; for (int q = 0; q < m; ++q) p *= RHO; }
        if (dk == 0) wv0 = p; else wv1 = p;
      }
      v2f tmp = {wv0, wv1};
      Bw[c] = tmp;
    }
    const int arow = lane & 15;           // A row = batch row within tile
    for (int st = wave; st < 17; st += 8) {   // 17 subtiles cover M cols [0, 271]
      v8f acc = {0.f, 0.f, 0.f, 0.f, 0.f, 0.f, 0.f, 0.f};
      const int jb = 16 * st;
#pragma unroll
      for (int c = 0; c < 12; ++c) {
        v2f a = *reinterpret_cast<const v2f*>(&b_s[arow * BS + jb + 4 * c + klo]);
        acc = __builtin_amdgcn_wmma_f32_16x16x4_f32(
            false, a, false, Bw[c], (short)0, acc, false, false);
      }
      const int mb = (lane >> 4) << 3;    // D rows: v + 8*(lane>=16)
#pragma unroll
      for (int v = 0; v < 8; ++v)
        m_s[(mb + v) * MS + jb + nn] = CINV * acc[v];
    }
  }
#else
  // VALU fallback: per-column 33-tap symmetric convolution
  for (int idx = tid; idx < 16 * 272; idx += 256) {
    int rr = idx / 272, col = idx - rr * 272;
    const float* br = &b_s[rr * BS + col];
    float w = 1.f, acc = br[16];
    for (int q = 1; q <= 16; ++q) { w *= RHO; acc += w * (br[16 - q] + br[16 + q]); }
    m_s[rr * MS + col] = CINV * acc;
  }
#endif
  __syncthreads();

  // ---- Phase 4: boundary images + not-a-knot endpoints (rank-1 corrections) ----
  if (c0 == 0 && tid < 16) {
    int r = tid;
    float SL = 0.f, pw = 1.f;                    // SL = sum_j rho^j b_j
    for (int j2 = 0; j2 <= 16; ++j2) { SL += pw * b_s[r * BS + 18 + j2]; pw *= RHO; }
    float pi = RHO * RHO;
    for (int i2 = 2; i2 <= 18; ++i2) { m_s[r * MS + i2] -= CINV * pi * SL; pi *= RHO; }
    float M1v = y_s[r * YS + 19] - 2.f * y_s[r * YS + 18] + y_s[r * YS + 17];
    m_s[r * MS + 1] = M1v;
    m_s[r * MS + 0] = 2.f * M1v - m_s[r * MS + 2];
  }
  if (c0 == (SPL_W - TILE) && tid >= 16 && tid < 32) {
    int r = tid - 16;
    int ie = SPL_W - 1 - c0 + 17;
    float Mn2 = y_s[r * YS + ie] - 2.f * y_s[r * YS + ie - 1] + y_s[r * YS + ie - 2];
    float SR = 0.f, pw = 1.f;                    // SR = sum_j rho^j b_{n-1-j}
    for (int j2 = 0; j2 <= 16; ++j2) { SR += pw * b_s[r * BS + (SPL_W - 5 - j2 - c0 + 18)]; pw *= RHO; }
    float pe = RHO * RHO;
    for (int e = 2; e <= 18; ++e) { m_s[r * MS + (SPL_W - 1 - e - c0)] -= CINV * pe * SR; pe *= RHO; }
    m_s[r * MS + (SPL_W - 2 - c0)] = Mn2;
    m_s[r * MS + (SPL_W - 1 - c0)] = 2.f * Mn2 - m_s[r * MS + (SPL_W - 3 - c0)];
  }
  __syncthreads();

  // ---- Phase 5: evaluate spline at xq = j*(w-1)/(w*r-1) for intervals in tile ----
  const float S    = 8191.0f / 32767.0f;
  const float INVS = 32767.0f / 8191.0f;
  int jstart = (int)((float)c0 * INVS) - 2; if (jstart < 0) jstart = 0;
  int jend   = (int)((float)(c0 + TILE) * INVS) + 3;
  if (jend > SPL_W * SPL_R) jend = SPL_W * SPL_R;
  for (int rr = 0; rr < 16; ++rr) {
    const float* yr = &y_s[rr * YS];
    const float* mr = &m_s[rr * MS];
    float* orow = &out[(size_t)(row0 + rr) * (SPL_W * SPL_R)];
    for (int j = jstart + tid; j < jend; j += 256) {
      float xq = (float)j * S;
      int i = (int)xq;
      if (i > SPL_W - 2) i = SPL_W - 2;
      if (i < c0 || i >= c0 + TILE) continue;    // exact ownership test, no overlap
      float t  = xq - (float)i;
      int  li  = i - c0;
      float y0 = yr[li + 17], y1 = yr[li + 18];
      float M0 = mr[li], M1 = mr[li + 1];
      float bb  = y1 - y0 - (2.f * M0 + M1) * (1.f / 6.f);
      float val = y0 + t * (bb + t * (0.5f * M0 + t * (M1 - M0) * (1.f / 6.f)));
      orow[j] = val;
    }
  }
}

extern "C" void kernel_launch(void* const* d_in, const int* in_sizes, int n_in,
                              void* d_out, int out_size, void* d_ws, size_t ws_size,
                              hipStream_t stream) {
  (void)in_sizes; (void)n_in; (void)out_size; (void)d_ws; (void)ws_size;
  const float* x = (const float*)d_in[0];
  float* out = (float*)d_out;
  dim3 grid(SPL_W / TILE, SPL_B / 16);   // 32 x 32 blocks
  Spline_63977832841689_kernel<<<grid, 256, 0, stream>>>(x, out);
}